// VMGLinear_27504970564168
// MI455X (gfx1250) — compile-verified
//
#include <hip/hip_runtime.h>
#include <cstdint>
#include <cstddef>

// ---------------------------------------------------------------------------
// VMGLinear for MI455X (gfx1250, wave32).
//   h[b,o] = (x1 @ mu)[b,o] + s_c[o] * sum_i u[b,i] * E[b,i,o]
//   u[b,i] = x1[b,i] * s_r[i],  s_r = sqrt(exp(logvar_in)), s_c likewise.
// E-stream (1.07 GB, touched once) is the HBM-bound part -> float4 streaming.
// x1@mu is a shared-operand fp32 GEMM -> V_WMMA_F32_16X16X4_F32 (exact).
// ---------------------------------------------------------------------------

typedef __attribute__((ext_vector_type(2))) float v2f;
typedef __attribute__((ext_vector_type(8))) float v8f;

#define MB       256
#define IN_DIM   1025   // includes bias row
#define OUT_DIM  1024
#define U_STRIDE 1028   // pad 1025 -> multiple of 4 for aligned float4 loads

// ---------------- D_KL: single-block global reductions ----------------------
__device__ __forceinline__ float block_reduce_1024(float v, float* red) {
    const int t = threadIdx.x;
    red[t] = v;
    __syncthreads();
    for (int s = 512; s > 0; s >>= 1) {
        if (t < s) red[t] += red[t + s];
        __syncthreads();
    }
    float r = red[0];
    __syncthreads();
    return r;
}

__global__ void vmg_dkl_kernel(const float* __restrict__ mu,
                               const float* __restrict__ lv_in,
                               const float* __restrict__ lv_out,
                               float* __restrict__ dkl_out) {
    __shared__ float red[1024];
    const int t = threadIdx.x;

    float s_var_r = 0.f, s_lv_in = 0.f;
    for (int i = t; i < IN_DIM; i += 1024) {
        float l = lv_in[i];
        s_var_r += expf(l);
        s_lv_in += l;
    }
    float l_out = lv_out[t];            // OUT_DIM == blockDim == 1024
    float s_var_c = expf(l_out);
    float s_lv_out = l_out;

    float s_mumu = 0.f;
    for (int idx = t; idx < IN_DIM * OUT_DIM; idx += 1024) {
        float m = mu[idx];
        s_mumu = fmaf(m, m, s_mumu);
    }

    s_var_r  = block_reduce_1024(s_var_r,  red);
    s_lv_in  = block_reduce_1024(s_lv_in,  red);
    s_var_c  = block_reduce_1024(s_var_c,  red);
    s_lv_out = block_reduce_1024(s_lv_out, red);
    s_mumu   = block_reduce_1024(s_mumu,   red);

    if (t == 0) {
        const float r = (float)IN_DIM, c = (float)OUT_DIM;
        *dkl_out = 0.5f * (s_var_r * s_var_c + s_mumu
                           - r * c - c * s_lv_in - r * s_lv_out);
    }
}

// ---------------- u[b,i] = x1[b,i] * sqrt(exp(lv_in[i])) --------------------
__global__ void vmg_u_kernel(const float* __restrict__ x,
                             const float* __restrict__ lv_in,
                             float* __restrict__ u) {
    int idx = blockIdx.x * 256 + threadIdx.x;
    if (idx >= MB * IN_DIM) return;
    int b = idx / IN_DIM;
    int i = idx - b * IN_DIM;
    float sr = sqrtf(expf(lv_in[i]));
    float xv = (i < IN_DIM - 1) ? x[(size_t)b * (IN_DIM - 1) + i] : 1.0f;
    u[(size_t)b * U_STRIDE + i] = xv * sr;
}

// ---------------- hgemm = x1 @ mu via V_WMMA_F32_16X16X4_F32 ----------------
// One wave32 per 16x16 tile. A(16x4): lanes0-15 M=0..15 hold K={0,1},
// lanes16-31 hold K={2,3}. B(4x16): VGPR0 lanes0-15 K=0 / lanes16-31 K=2,
// VGPR1 K=1/K=3. C(16x16): VGPRv lanes0-15 M=v, lanes16-31 M=v+8; N=lane%16.
__global__ void vmg_gemm_wmma(const float* __restrict__ x,
                              const float* __restrict__ mu,
                              float* __restrict__ hg) {
    const int lane = threadIdx.x;        // 0..31
    const int half = lane >> 4;          // 0/1
    const int lm   = lane & 15;
    const int m0   = blockIdx.y * 16;
    const int n0   = blockIdx.x * 16;
    const int n    = n0 + lm;

    const float* xrow = x + (size_t)(m0 + lm) * (IN_DIM - 1);

    v8f c = {};
#pragma unroll 4
    for (int k = 0; k < IN_DIM - 1; k += 4) {
        const int ka = k + 2 * half;
        v2f a, b;
        a.x = xrow[ka];
        a.y = xrow[ka + 1];
        b.x = mu[(size_t)ka * OUT_DIM + n];
        b.y = mu[(size_t)(ka + 1) * OUT_DIM + n];
        c = __builtin_amdgcn_wmma_f32_16x16x4_f32(
                /*neg_a=*/false, a, /*neg_b=*/false, b,
                /*c_mod=*/(short)0, c, /*reuse_a=*/false, /*reuse_b=*/false);
    }

    // bias row of x1 (== 1.0) contributes mu[1024, n] to every output row
    const float bias = mu[(size_t)(IN_DIM - 1) * OUT_DIM + n];

#pragma unroll
    for (int v = 0; v < 8; ++v) {
        const int row = m0 + v + 8 * half;
        hg[(size_t)row * OUT_DIM + n] = c[v] + bias;
    }
}

// ---------------- E-stream: y[b,o] = sum_i u[b,i] * E[b,i,o]; fuse epilogue -
__device__ __forceinline__ void fma4(float4& acc, float s, const float4 e) {
    acc.x = fmaf(s, e.x, acc.x);
    acc.y = fmaf(s, e.y, acc.y);
    acc.z = fmaf(s, e.z, acc.z);
    acc.w = fmaf(s, e.w, acc.w);
}

__global__ void vmg_evm_kernel(const float* __restrict__ E,
                               const float* __restrict__ lv_out,
                               const float* __restrict__ u,
                               const float* __restrict__ hg,
                               float* __restrict__ out) {
    const int b  = blockIdx.y;
    const int o4 = blockIdx.x * 512 + threadIdx.x * 4;   // 4 consecutive cols

    const float* __restrict__ Eb = E + (size_t)b * IN_DIM * OUT_DIM;
    const float* __restrict__ ub = u + (size_t)b * U_STRIDE;

    float4 acc = make_float4(0.f, 0.f, 0.f, 0.f);

    for (int i = 0; i < IN_DIM - 1; i += 4) {
        // keep the stream ahead of us in flight (speculative, OOB-safe)
        __builtin_prefetch(Eb + (size_t)(i + 16) * OUT_DIM + o4, 0, 1);
        const float4 uv = *(const float4*)(ub + i);           // 16B aligned
        const float4 e0 = *(const float4*)(Eb + (size_t)(i + 0) * OUT_DIM + o4);
        const float4 e1 = *(const float4*)(Eb + (size_t)(i + 1) * OUT_DIM + o4);
        const float4 e2 = *(const float4*)(Eb + (size_t)(i + 2) * OUT_DIM + o4);
        const float4 e3 = *(const float4*)(Eb + (size_t)(i + 3) * OUT_DIM + o4);
        fma4(acc, uv.x, e0);
        fma4(acc, uv.y, e1);
        fma4(acc, uv.z, e2);
        fma4(acc, uv.w, e3);
    }
    { // tail: i = 1024 (the bias row of E)
        const float ul   = ub[IN_DIM - 1];
        const float4 et  = *(const float4*)(Eb + (size_t)(IN_DIM - 1) * OUT_DIM + o4);
        fma4(acc, ul, et);
    }

    const float4 h = *(const float4*)(hg + (size_t)b * OUT_DIM + o4);
    float4 r;
    r.x = fmaf(sqrtf(expf(lv_out[o4 + 0])), acc.x, h.x);
    r.y = fmaf(sqrtf(expf(lv_out[o4 + 1])), acc.y, h.y);
    r.z = fmaf(sqrtf(expf(lv_out[o4 + 2])), acc.z, h.z);
    r.w = fmaf(sqrtf(expf(lv_out[o4 + 3])), acc.w, h.w);
    *(float4*)(out + (size_t)b * OUT_DIM + o4) = r;
}

// ---------------------------------------------------------------------------
extern "C" void kernel_launch(void* const* d_in, const int* in_sizes, int n_in,
                              void* d_out, int out_size, void* d_ws, size_t ws_size,
                              hipStream_t stream) {
    const float* x      = (const float*)d_in[0];   // (256, 1024)
    const float* mu     = (const float*)d_in[1];   // (1025, 1024)
    const float* lv_in  = (const float*)d_in[2];   // (1025,)
    const float* lv_out = (const float*)d_in[3];   // (1024,)
    const float* E      = (const float*)d_in[4];   // (256, 1025, 1024)
    float* out = (float*)d_out;                    // h (256*1024) then D_KL

    float* u  = (float*)d_ws;                      // 256 * 1028 floats
    float* hg = u + (size_t)MB * U_STRIDE;         // 256 * 1024 floats

    // 1) D_KL (independent reductions)
    vmg_dkl_kernel<<<1, 1024, 0, stream>>>(mu, lv_in, lv_out,
                                           out + (size_t)MB * OUT_DIM);
    // 2) u = x1 * s_r
    vmg_u_kernel<<<(MB * IN_DIM + 255) / 256, 256, 0, stream>>>(x, lv_in, u);
    // 3) hgemm = x1 @ mu  (WMMA f32, one wave per 16x16 tile)
    vmg_gemm_wmma<<<dim3(OUT_DIM / 16, MB / 16), 32, 0, stream>>>(x, mu, hg);
    // 4) stream E, fuse  out = hgemm + s_c * (u . E)
    vmg_evm_kernel<<<dim3(2, MB), 128, 0, stream>>>(E, lv_out, u, hg, out);
}